// Fusedmax_23106924052891
// MI455X (gfx1250) — compile-verified
//
#include <hip/hip_runtime.h>
#include <stdint.h>

// Fusedmax (TV1D-prox + sparsemax), MI455X / gfx1250.
// One thread per row; rows staged into LDS by the Tensor Data Mover with
// descriptor-side padding (+1 dword per 256) so the row stride is 514 dwords
// -> bank-conflict-free lockstep access across the wave32 lanes.

#define ALPHAF 0.1f
#define NCOLS 512
#define ROW_STRIDE 514                    // 512 data dwords + 2 TDM pad dwords
#define ROWS_PER_WAVE 32
#define WAVES_PER_BLOCK 2
#define ROWS_PER_BLOCK (ROWS_PER_WAVE * WAVES_PER_BLOCK)   // 64
#define LDS_FLOATS (ROWS_PER_BLOCK * ROW_STRIDE)           // 32896
#define SMEM_BYTES ((LDS_FLOATS + ROWS_PER_BLOCK) * 4)     // rows + taus

typedef __attribute__((ext_vector_type(4))) unsigned int u32x4;
typedef __attribute__((ext_vector_type(8))) int i32x8;
typedef __attribute__((ext_vector_type(4))) int i32x4;

#if defined(__has_builtin)
#if __has_builtin(__builtin_amdgcn_tensor_load_to_lds)
#define HAVE_TDM 1
#endif
#endif
#ifndef HAVE_TDM
#define HAVE_TDM 0
#endif

// LDS index of column c inside a padded row: hole at dword 256 (and 513),
// matching the TDM pad placement (pad_interval=256 dwords, pad_amount=1).
__device__ __forceinline__ int pidx(int c) { return c + (c >> 8); }

__global__ __launch_bounds__(ROWS_PER_BLOCK)
void fusedmax_kernel(const float* __restrict__ x, float* __restrict__ out, int nRows) {
  extern __shared__ float smem[];
  float* taus = smem + LDS_FLOATS;

  const int lane     = threadIdx.x & 31;
  const int waveId   = threadIdx.x >> 5;
  const int waveRow0 = blockIdx.x * ROWS_PER_BLOCK + waveId * ROWS_PER_WAVE;
  float* waveLds = smem + (size_t)waveId * ROWS_PER_WAVE * ROW_STRIDE;

  int waveRows = nRows - waveRow0;
  if (waveRows > ROWS_PER_WAVE) waveRows = ROWS_PER_WAVE;

  // ---------------- stage tile: global -> LDS ----------------
  if (waveRows > 0) {
#if HAVE_TDM
    // Tensor DMA descriptor (D#), 2-D tile [waveRows x 512] of 4-byte data.
    uint64_t ga = (uint64_t)(uintptr_t)(x + (size_t)waveRow0 * NCOLS);
    uint32_t lb = (uint32_t)(uintptr_t)waveLds;   // low 32 bits = LDS byte offset

    u32x4 g0;
    g0.x = 1u;                                              // count=1 (valid), user mode
    g0.y = lb;                                              // lds_addr (bytes)
    g0.z = (uint32_t)ga;                                    // global_addr[31:0]
    g0.w = (uint32_t)((ga >> 32) & 0x01FFFFFFu)             // global_addr[56:32]
         | 0x80000000u;                                     // type=2 ("image")

    i32x8 g1;
    g1[0] = (int)((2u << 16)      // data_size = 4B
                | (1u << 20)      // pad_enable
                | (7u << 22)      // pad_interval code 7 = every 256 dwords
                | (0u << 25));    // pad_amount  code 0 = 1 dword
    g1[1] = (int)((uint32_t)NCOLS << 16);       // tensor_dim0[15:0] @ [79:64]... bits[31:16]
    g1[2] = (int)((uint32_t)waveRows << 16);    // tensor_dim1[15:0] in bits[31:16]
    g1[3] = (int)((uint32_t)NCOLS << 16);       // tile_dim0 in bits[31:16]
    g1[4] = waveRows;                           // tile_dim1 (tile_dim2 = 0)
    g1[5] = NCOLS;                              // tensor_dim0_stride[31:0] (elements)
    g1[6] = 0;                                  // stride hi / tensor_dim1_stride lo
    g1[7] = 0;

    i32x4 gz = {0, 0, 0, 0};                    // groups 2/3: dims 2..4 unused
#if __has_include(<hip/amd_detail/amd_gfx1250_TDM.h>)
    i32x8 gz8 = {0, 0, 0, 0, 0, 0, 0, 0};
    __builtin_amdgcn_tensor_load_to_lds(g0, g1, gz, gz, gz8, 0);
#else
    __builtin_amdgcn_tensor_load_to_lds(g0, g1, gz, gz, 0);
#endif
    __builtin_amdgcn_s_wait_tensorcnt(0);
    asm volatile("" ::: "memory");
#else
    for (int r = 0; r < waveRows; ++r)
      for (int c = lane; c < NCOLS; c += 32)
        waveLds[r * ROW_STRIDE + pidx(c)] = x[(size_t)(waveRow0 + r) * NCOLS + c];
    asm volatile("" ::: "memory");
#endif
  }

  // ---------------- per-thread row: TV1D prox (Condat, in-place) ----------------
  const int myRow = waveRow0 + lane;
  if (myRow < nRows) {
    float* row = waveLds + lane * ROW_STRIDE;
    const float lam = ALPHAF;
    const int n = NCOLS;

    int k = 0, k0 = 0, km = 0, kp = 0;
    float w0v  = row[0];
    float vmin = w0v - lam, vmax = w0v + lam;
    float umin = lam, umax = -lam;

    for (int it = 0; it < 40 * NCOLS; ++it) {
      if (k >= n - 1) {
        if (umin < 0.f) {                                   // A: flush vmin segment
          for (int i = k0; i <= km && i < n; ++i) row[pidx(i)] = vmin;
          int k0n = km + 1;
          float wk0 = row[pidx(k0n < n ? k0n : n - 1)];
          k = k0n; k0 = k0n; km = k0n;                      // kp, vmax unchanged
          vmin = wk0; umin = lam;
          umax = wk0 + lam - vmax;
        } else if (umax > 0.f) {                            // B: flush vmax segment
          for (int i = k0; i <= kp && i < n; ++i) row[pidx(i)] = vmax;
          int k0n = kp + 1;
          float wk0 = row[pidx(k0n < n ? k0n : n - 1)];
          k = k0n; k0 = k0n; kp = k0n;                      // km, vmin unchanged
          vmax = wk0; umax = -lam;
          umin = wk0 - lam - vmin;
        } else {                                            // C: final segment, done
          float v = vmin + umin / (float)(k - k0 + 1);
          for (int i = k0; i < n; ++i) row[pidx(i)] = v;
          break;
        }
      } else {
        float wk1 = row[pidx(k + 1)];
        float umin1 = umin + wk1 - vmin;
        float umax1 = umax + wk1 - vmax;
        if (umin1 < -lam) {                                 // D: negative jump
          for (int i = k0; i <= km && i < n; ++i) row[pidx(i)] = vmin;
          int k0n = km + 1;
          float wk0 = row[pidx(k0n)];
          k = k0n; k0 = k0n; km = k0n; kp = k0n;
          vmin = wk0; vmax = wk0 + 2.f * lam;
          umin = lam; umax = -lam;
        } else if (umax1 > lam) {                           // E: positive jump
          for (int i = k0; i <= kp && i < n; ++i) row[pidx(i)] = vmax;
          int k0n = kp + 1;
          float wk0 = row[pidx(k0n)];
          k = k0n; k0 = k0n; km = k0n; kp = k0n;
          vmax = wk0; vmin = wk0 - 2.f * lam;
          umin = lam; umax = -lam;
        } else {                                            // F: advance
          int kF = k + 1;
          float den = (float)(kF - k0 + 1);
          if (umin1 >= lam) { vmin += (umin1 - lam) / den; umin = lam; km = kF; }
          else              { umin = umin1; }
          if (umax1 <= -lam){ vmax += (umax1 + lam) / den; umax = -lam; kp = kF; }
          else              { umax = umax1; }
          k = kF;
        }
      }
    }

    // ---------------- sparsemax threshold via bisection ----------------
    // f(tau) = sum(max(z - tau, 0)) is monotone piecewise-linear; root in
    // [zmax - 1, zmax]. 34 halvings >> fp32 resolution -> exact to ulp.
    float zmax = -3.402823466e38f;
    for (int c = 0; c < NCOLS; ++c) {
      float v = row[pidx(c)];
      zmax = v > zmax ? v : zmax;
    }
    float lo = zmax - 1.0f, hi = zmax;
    for (int it = 0; it < 34; ++it) {
      float mid = 0.5f * (lo + hi);
      float s = 0.f;
      for (int c = 0; c < NCOLS; ++c) {
        float d = row[pidx(c)] - mid;
        s += d > 0.f ? d : 0.f;
      }
      if (s >= 1.0f) lo = mid; else hi = mid;
    }
    taus[threadIdx.x] = 0.5f * (lo + hi);
  }

  __syncthreads();

  // ---------------- coalesced output: lane-cooperative over the wave's rows ----
  for (int r = 0; r < waveRows; ++r) {
    float t = taus[waveId * ROWS_PER_WAVE + r];
    const float* rrow = waveLds + (size_t)r * ROW_STRIDE;
    float* gout = out + (size_t)(waveRow0 + r) * NCOLS;
    for (int c = lane; c < NCOLS; c += 32) {
      float v = rrow[pidx(c)] - t;
      gout[c] = v > 0.f ? v : 0.f;
    }
  }
}

extern "C" void kernel_launch(void* const* d_in, const int* in_sizes, int n_in,
                              void* d_out, int out_size, void* d_ws, size_t ws_size,
                              hipStream_t stream) {
  (void)n_in; (void)out_size; (void)d_ws; (void)ws_size;
  const float* x = (const float*)d_in[0];
  float* out = (float*)d_out;
  int nRows = in_sizes[0] / NCOLS;
  int grid = (nRows + ROWS_PER_BLOCK - 1) / ROWS_PER_BLOCK;
  fusedmax_kernel<<<grid, ROWS_PER_BLOCK, SMEM_BYTES, stream>>>(x, out, nRows);
}